// TemporalGNN_14096082665689
// MI455X (gfx1250) — compile-verified
//
#include <hip/hip_runtime.h>
#include <math.h>

typedef __attribute__((ext_vector_type(2))) float v2f;
typedef __attribute__((ext_vector_type(8))) float v8f;

constexpr int kN = 50000;   // nodes
constexpr int kF = 8;       // input features
constexpr int kP = 12;      // periods
constexpr int kH = 64;      // hidden
constexpr int kE = 800000;  // edges

// ---------------------------------------------------------------------------
// V_WMMA_F32_16X16X4_F32 : D(16x16) = A(16x4) * B(4x16) + C   (full fp32)
// ---------------------------------------------------------------------------
static __device__ __forceinline__ v8f wmma4(v2f a, v2f b, v8f c) {
  return __builtin_amdgcn_wmma_f32_16x16x4_f32(false, a, false, b, (short)0, c,
                                               false, false);
}

// Branch-free fast activations (v_exp_f32 + v_rcp_f32; no EXEC divergence).
static __device__ __forceinline__ float fast_sigmoid(float x) {
  float e = __expf(-x);  // overflow -> inf -> rcp -> 0 (correct limit)
  return __builtin_amdgcn_rcpf(1.0f + e);
}
static __device__ __forceinline__ float fast_tanh(float x) {
  float e = __expf(-2.0f * fabsf(x));  // e in (0, 1], never overflows
  float t = (1.0f - e) * __builtin_amdgcn_rcpf(1.0f + e);
  return copysignf(t, x);
}

// ---------------------------------------------------------------------------
// scratch zeroing (grid-stride)
// ---------------------------------------------------------------------------
__global__ void zero_kernel(float* __restrict__ p, long n) {
  long i = (long)blockIdx.x * blockDim.x + threadIdx.x;
  long stride = (long)gridDim.x * blockDim.x;
  for (; i < n; i += stride) p[i] = 0.0f;
}

// ---------------------------------------------------------------------------
// Fold params: Mz = conv_w[0] @ lin_w[0][:64]  (8x64), cz = conv_b[0]@Lz+lin_b[0]
//              Mh = conv_w[2] @ lin_w[2][:64],         ch = conv_b[2]@Lh+lin_b[2]
//              probs = softmax(attention)
// ---------------------------------------------------------------------------
__global__ __launch_bounds__(64) void params_kernel(
    const float* __restrict__ att, const float* __restrict__ conv_w,
    const float* __restrict__ conv_b, const float* __restrict__ lin_w,
    const float* __restrict__ lin_b, float* __restrict__ Mz,
    float* __restrict__ Mh, float* __restrict__ cz, float* __restrict__ ch,
    float* __restrict__ probs) {
  const int c = threadIdx.x;
  if (c == 0) {
    float m = -1e30f;
    for (int p = 0; p < kP; ++p) m = fmaxf(m, att[p]);
    float s = 0.0f, e[kP];
    for (int p = 0; p < kP; ++p) { e[p] = __expf(att[p] - m); s += e[p]; }
    float inv = 1.0f / s;
    for (int p = 0; p < kP; ++p) probs[p] = e[p] * inv;
  }
  const float* Lz = lin_w;                 // lin_w[0], rows 0..63 used
  const float* Lh = lin_w + 2 * 128 * 64;  // lin_w[2]
  for (int i = 0; i < kF; ++i) {
    float sz = 0.0f, sh = 0.0f;
    for (int j = 0; j < kH; ++j) {
      sz += conv_w[0 * kF * kH + i * kH + j] * Lz[j * kH + c];
      sh += conv_w[2 * kF * kH + i * kH + j] * Lh[j * kH + c];
    }
    Mz[i * kH + c] = sz;
    Mh[i * kH + c] = sh;
  }
  float sz = 0.0f, sh = 0.0f;
  for (int j = 0; j < kH; ++j) {
    sz += conv_b[j] * Lz[j * kH + c];
    sh += conv_b[2 * kH + j] * Lh[j * kH + c];
  }
  cz[c] = sz + lin_b[c];
  ch[c] = sh + lin_b[2 * kH + c];
}

// ---------------------------------------------------------------------------
// degree = segment_sum(w, col)
// ---------------------------------------------------------------------------
__global__ void deg_kernel(const int* __restrict__ ei,
                           const float* __restrict__ ew,
                           float* __restrict__ deg) {
  int e = blockIdx.x * blockDim.x + threadIdx.x;
  if (e < kE) atomicAdd(&deg[ei[kE + e]], ew[e]);
}

// deg -> dinv = rsqrt(deg + 1)  (in place; deg+1 > 0 always)
__global__ void dinv_kernel(float* __restrict__ deg) {
  int n = blockIdx.x * blockDim.x + threadIdx.x;
  if (n < kN) deg[n] = rsqrtf(deg[n] + 1.0f);
}

__global__ void enorm_kernel(const int* __restrict__ ei,
                             const float* __restrict__ ew,
                             const float* __restrict__ dinv,
                             float* __restrict__ enorm) {
  int e = blockIdx.x * blockDim.x + threadIdx.x;
  if (e < kE) enorm[e] = dinv[ei[e]] * ew[e] * dinv[ei[kE + e]];
}

// ax[n][p][f] = self_norm[n] * x[n][f][p]   (also transposes (F,P)->(P,F))
__global__ void selfinit_kernel(const float* __restrict__ x,
                                const float* __restrict__ dinv,
                                float* __restrict__ ax) {
  long i = (long)blockIdx.x * blockDim.x + threadIdx.x;
  if (i < (long)kN * kF * kP) {
    int n = (int)(i / 96);
    int k = (int)(i % 96);
    int p = k >> 3, f = k & 7;
    float d = dinv[n];
    ax[i] = d * d * x[(long)n * 96 + f * kP + p];
  }
}

// ax[col] += enorm[e] * x[row]   (all 12 periods x 8 features per edge)
__global__ void scatter_kernel(const int* __restrict__ ei,
                               const float* __restrict__ x,
                               const float* __restrict__ enorm,
                               float* __restrict__ ax) {
  long t = (long)blockIdx.x * blockDim.x + threadIdx.x;
  if (t < (long)kE * 96) {
    int e = (int)(t / 96);
    int k = (int)(t % 96);
    int p = k >> 3, f = k & 7;
    int r = ei[e], c = ei[kE + e];
    atomicAdd(&ax[(long)c * 96 + k], enorm[e] * x[(long)r * 96 + f * kP + p]);
  }
}

// ---------------------------------------------------------------------------
// Gate kernel: one wave = 16 nodes.
//   per period: Z = sigmoid(ax@Mz+cz), Ht = tanh(ax@Mh+ch), H += prob*(1-Z)*Ht
//   then Hrelu = relu(H) stored (N,64) + per-column sum/sumsq for BatchNorm.
// Biases are injected as the WMMA C operand (per-lane column is constant).
// ---------------------------------------------------------------------------
__global__ __launch_bounds__(32) void gate_kernel(
    const float* __restrict__ ax, const float* __restrict__ Mz,
    const float* __restrict__ Mh, const float* __restrict__ cz,
    const float* __restrict__ ch, const float* __restrict__ probs,
    float* __restrict__ Hrelu, float* __restrict__ sums) {
  const int lane = threadIdx.x;
  const int half = lane >> 4;   // 0 or 1
  const int nrow = lane & 15;   // M (A) / N (B,C) index
  const int base = blockIdx.x * 16;
  const int node = base + nrow;

  // Preload B operands: 4 N-tiles x 2 K-steps, both gates; bias as C-init.
  v2f Bz[4][2], Bh[4][2];
  v8f Cz[4], Ch[4];
#pragma unroll
  for (int t = 0; t < 4; ++t) {
    const int col = t * 16 + nrow;
#pragma unroll
    for (int s = 0; s < 2; ++s) {
      const int k0 = s * 4 + half * 2;
      Bz[t][s].x = Mz[k0 * kH + col];
      Bz[t][s].y = Mz[(k0 + 1) * kH + col];
      Bh[t][s].x = Mh[k0 * kH + col];
      Bh[t][s].y = Mh[(k0 + 1) * kH + col];
    }
    const float czv = cz[col], chv = ch[col];
#pragma unroll
    for (int r = 0; r < 8; ++r) { Cz[t][r] = czv; Ch[t][r] = chv; }
  }

  v8f Hacc[4] = {};
  for (int p = 0; p < kP; ++p) {
    const float pr = probs[p];
    const float* axp = ax + (long)node * 96 + p * 8;
    v2f a0, a1;
    a0.x = axp[half * 2];     a0.y = axp[half * 2 + 1];
    a1.x = axp[4 + half * 2]; a1.y = axp[4 + half * 2 + 1];
#pragma unroll
    for (int t = 0; t < 4; ++t) {
      v8f az = wmma4(a0, Bz[t][0], Cz[t]);
      az = wmma4(a1, Bz[t][1], az);
      v8f ah = wmma4(a0, Bh[t][0], Ch[t]);
      ah = wmma4(a1, Bh[t][1], ah);
#pragma unroll
      for (int r = 0; r < 8; ++r) {
        float z = fast_sigmoid(az[r]);
        float h = fast_tanh(ah[r]);
        Hacc[t][r] += pr * (1.0f - z) * h;
      }
    }
  }

  // relu + store + BN partial sums
#pragma unroll
  for (int t = 0; t < 4; ++t) {
    const int col = t * 16 + nrow;
    float s = 0.0f, q = 0.0f;
#pragma unroll
    for (int r = 0; r < 8; ++r) {
      float v = fmaxf(Hacc[t][r], 0.0f);
      s += v;
      q += v * v;
      Hrelu[(long)(base + r + 8 * half) * kH + col] = v;
    }
    s += __shfl_xor(s, 16);  // fold the two lane-halves (same column)
    q += __shfl_xor(q, 16);
    if (half == 0) {
      atomicAdd(&sums[col], s);
      atomicAdd(&sums[kH + col], q);
    }
  }
}

// BatchNorm finalize: scale/shift per feature
__global__ __launch_bounds__(64) void bn_kernel(const float* __restrict__ sums,
                                                const float* __restrict__ gamma,
                                                const float* __restrict__ beta,
                                                float* __restrict__ scale,
                                                float* __restrict__ shift) {
  const int c = threadIdx.x;
  const float invN = 1.0f / (float)kN;
  float mean = sums[c] * invN;
  float var = sums[kH + c] * invN - mean * mean;
  float sc = gamma[c] * rsqrtf(var + 1e-5f);
  scale[c] = sc;
  shift[c] = beta[c] - mean * sc;
}

// ---------------------------------------------------------------------------
// MLP: one wave = 16 nodes; intermediates staged through LDS for the D-tile ->
// A-operand layout conversion between WMMA layers. Bias rides in C operand.
// ---------------------------------------------------------------------------
template <int K, int Nc>
__device__ __forceinline__ void lds_gemm(const float (&in)[16][96],
                                         const float* __restrict__ W,
                                         const float* __restrict__ bias,
                                         float (&outb)[16][96], int lane) {
  const int half = lane >> 4, nrow = lane & 15;
  constexpr int NT = (Nc + 15) / 16;
#pragma unroll
  for (int t = 0; t < NT; ++t) {
    const int col = t * 16 + nrow;
    const float bv = (col < Nc) ? bias[col] : 0.0f;
    v8f acc;
#pragma unroll
    for (int r = 0; r < 8; ++r) acc[r] = bv;
#pragma unroll
    for (int s = 0; s < K / 4; ++s) {
      const int k0 = s * 4 + half * 2;
      v2f a, b;
      a.x = in[nrow][k0];
      a.y = in[nrow][k0 + 1];
      b.x = (col < Nc) ? W[k0 * Nc + col] : 0.0f;
      b.y = (col < Nc) ? W[(k0 + 1) * Nc + col] : 0.0f;
      acc = wmma4(a, b, acc);
    }
    if (col < Nc) {
#pragma unroll
      for (int r = 0; r < 8; ++r) outb[r + 8 * half][col] = acc[r];
    }
  }
}

__global__ __launch_bounds__(32) void mlp_kernel(
    const float* __restrict__ Hrelu, const float* __restrict__ scale,
    const float* __restrict__ shift, const float* __restrict__ W1,
    const float* __restrict__ b1, const float* __restrict__ W2,
    const float* __restrict__ b2, const float* __restrict__ W3,
    const float* __restrict__ b3, const float* __restrict__ W4,
    const float* __restrict__ b4, const float* __restrict__ W5,
    const float* __restrict__ b5, float* __restrict__ out) {
  __shared__ float bufA[16][96];
  __shared__ float bufB[16][96];
  const int lane = threadIdx.x;
  const int half = lane >> 4, nrow = lane & 15;
  const int base = blockIdx.x * 16;

  // layer-0 input with BatchNorm applied
  for (int i = lane; i < 16 * 64; i += 32) {
    int r = i >> 6, c = i & 63;
    bufA[r][c] = Hrelu[(long)(base + r) * kH + c] * scale[c] + shift[c];
  }
  __syncthreads();
  lds_gemm<64, 64>(bufA, W1, b1, bufB, lane);
  __syncthreads();
  lds_gemm<64, 32>(bufB, W2, b2, bufA, lane);
  __syncthreads();
  lds_gemm<32, 16>(bufA, W3, b3, bufB, lane);
  __syncthreads();
  lds_gemm<16, 8>(bufB, W4, b4, bufA, lane);
  __syncthreads();

  // final layer (K=8, Nc=96) -> global, matches reshape(N, 8, 12) flat layout
#pragma unroll
  for (int t = 0; t < 6; ++t) {
    const int col = t * 16 + nrow;
    const float bv = b5[col];
    v8f acc;
#pragma unroll
    for (int r = 0; r < 8; ++r) acc[r] = bv;
#pragma unroll
    for (int s = 0; s < 2; ++s) {
      const int k0 = s * 4 + half * 2;
      v2f a, b;
      a.x = bufA[nrow][k0];
      a.y = bufA[nrow][k0 + 1];
      b.x = W5[k0 * 96 + col];
      b.y = W5[(k0 + 1) * 96 + col];
      acc = wmma4(a, b, acc);
    }
#pragma unroll
    for (int r = 0; r < 8; ++r)
      out[(long)(base + r + 8 * half) * 96 + col] = acc[r];
  }
}

// ---------------------------------------------------------------------------
extern "C" void kernel_launch(void* const* d_in, const int* in_sizes, int n_in,
                              void* d_out, int out_size, void* d_ws,
                              size_t ws_size, hipStream_t stream) {
  const float* x = (const float*)d_in[0];
  const int* ei = (const int*)d_in[1];
  const float* ew = (const float*)d_in[2];
  const float* att = (const float*)d_in[3];
  const float* conv_w = (const float*)d_in[4];
  const float* conv_b = (const float*)d_in[5];
  const float* lin_w = (const float*)d_in[6];
  const float* lin_b = (const float*)d_in[7];
  const float* gamma = (const float*)d_in[8];
  const float* beta = (const float*)d_in[9];
  const float* W1 = (const float*)d_in[10];
  const float* b1 = (const float*)d_in[11];
  const float* W2 = (const float*)d_in[12];
  const float* b2 = (const float*)d_in[13];
  const float* W3 = (const float*)d_in[14];
  const float* b3 = (const float*)d_in[15];
  const float* W4 = (const float*)d_in[16];
  const float* b4 = (const float*)d_in[17];
  const float* W5 = (const float*)d_in[18];
  const float* b5 = (const float*)d_in[19];
  float* out = (float*)d_out;

  // workspace layout (floats); ~35.5 MB total
  float* ws = (float*)d_ws;
  float* deg = ws;                          // kN  (becomes dinv in place)
  float* enorm = deg + kN;                  // kE
  float* ax = enorm + kE;                   // kN*96
  float* Hrelu = ax + (long)kN * 96;        // kN*64
  float* sums = Hrelu + (long)kN * 64;      // 128 (sum | sumsq)
  float* Mz = sums + 128;                   // 512
  float* Mh = Mz + 512;                     // 512
  float* cz = Mh + 512;                     // 64
  float* chb = cz + 64;                     // 64
  float* probs = chb + 64;                  // 16
  float* scale = probs + 16;                // 64
  float* shift = scale + 64;                // 64

  const long zeroCnt = (long)kN + kE + (long)kN * 96;  // deg + enorm + ax
  zero_kernel<<<2048, 256, 0, stream>>>(ws, zeroCnt);
  zero_kernel<<<1, 128, 0, stream>>>(sums, 128);

  params_kernel<<<1, 64, 0, stream>>>(att, conv_w, conv_b, lin_w, lin_b, Mz, Mh,
                                      cz, chb, probs);
  deg_kernel<<<(kE + 255) / 256, 256, 0, stream>>>(ei, ew, deg);
  dinv_kernel<<<(kN + 255) / 256, 256, 0, stream>>>(deg);
  enorm_kernel<<<(kE + 255) / 256, 256, 0, stream>>>(ei, ew, deg, enorm);
  selfinit_kernel<<<(int)(((long)kN * 96 + 255) / 256), 256, 0, stream>>>(
      x, deg, ax);
  scatter_kernel<<<(int)(((long)kE * 96 + 255) / 256), 256, 0, stream>>>(
      ei, x, enorm, ax);
  gate_kernel<<<kN / 16, 32, 0, stream>>>(ax, Mz, Mh, cz, chb, probs, Hrelu,
                                          sums);
  bn_kernel<<<1, 64, 0, stream>>>(sums, gamma, beta, scale, shift);
  mlp_kernel<<<kN / 16, 32, 0, stream>>>(Hrelu, scale, shift, W1, b1, W2, b2,
                                         W3, b3, W4, b4, W5, b5, out);
}